// PPRPowerIteration_23510650978740
// MI455X (gfx1250) — compile-verified
//
#include <hip/hip_runtime.h>
#include <math.h>

// ---------------- problem constants ----------------
#define NNODES 3072
#define NH     4
#define NC     64
#define NHC    256          // NH*NC
#define NITER  4
#define NITERA 4
#define NIDX   512

typedef float v2f __attribute__((ext_vector_type(2)));
typedef float v8f __attribute__((ext_vector_type(8)));

__device__ __forceinline__ v8f wmma4(v2f a, v2f b, v8f c) {
  // D(16x16,f32) = A(16x4,f32) x B(4x16,f32) + C   -- CDNA5 V_WMMA_F32_16X16X4_F32
  return __builtin_amdgcn_wmma_f32_16x16x4_f32(false, a, false, b, (short)0, c,
                                               false, false);
}

// ================= generic fp32 WMMA GEMM =================
// D[M,Nc] = scale * (op(A)[M,K] @ B[K,Nc]) (+ diag*I)
//   op(A) = A     (A row-major [M,K], lda)       (TA=0)
//   op(A) = A^T   (A row-major [K,M], lda)       (TA=1)
// LOSS=1: instead of storing D, fuse r = v - Adj; e = exp(-r*r)*r; sum e*e
//         into loss_part[block] (deterministic tree reduce, no atomics).
// Requires K % 32 == 0 (all K here: 3072, 64, 256). Partial 128-tiles in M/N OK.
#define BM 128
#define BN 128
#define KB 32
#define KPAD (KB + 4)   // 36 floats = 144B rows, 8B-aligned for v2f LDS loads

template <bool TA, bool LOSS>
__global__ __launch_bounds__(256) void gemm_wmma(
    float* __restrict__ D, const float* __restrict__ A, const float* __restrict__ B,
    int M, int Nc, int K, int lda, int ldb, int ldd,
    const float* scale_ptr, float scale_const, const float* diag_ptr,
    long long aBatch, long long bBatch, long long dBatch,
    const float* __restrict__ Adj, int ldadj, float* __restrict__ loss_part) {
  __shared__ float As[2][BM][KPAD];   // As[buf][m][k]
  __shared__ float Bs[2][BN][KPAD];   // Bs[buf][n][k]  (B stored transposed)
  __shared__ float red[256];

  const int tid  = threadIdx.x;
  const int lane = tid & 31;
  const int wave = tid >> 5;
  const int z    = blockIdx.z;
  A += (long long)z * aBatch;
  B += (long long)z * bBatch;
  D += (long long)z * dBatch;

  const int m0  = blockIdx.y * BM;
  const int n0  = blockIdx.x * BN;
  const int wmL = (wave >> 2) * 64;  // 2 waves in M
  const int wnL = (wave & 3) * 32;   // 4 waves in N
  const bool active = ((m0 + wmL) < M) && ((n0 + wnL) < Nc);  // wave-uniform

  const float s  = scale_const * (scale_ptr ? scale_ptr[0] : 1.0f);
  const float dv = diag_ptr ? diag_ptr[0] : 0.0f;

  v8f acc[4][2];
  const v8f z8 = {0.f, 0.f, 0.f, 0.f, 0.f, 0.f, 0.f, 0.f};
#pragma unroll
  for (int i = 0; i < 4; i++)
#pragma unroll
    for (int j = 0; j < 2; j++) acc[i][j] = z8;

  const int nch = K / KB;
  float ra[16], rb[16];

  // ---- prologue: fetch + store chunk 0 ----
#pragma unroll
  for (int i = 0; i < 16; i++) {
    int e = tid + i * 256;
    if (TA) {
      int m = e & (BM - 1), k = e >> 7;
      ra[i] = ((m0 + m) < M) ? A[(long long)k * lda + (m0 + m)] : 0.f;
    } else {
      int m = e >> 5, k = e & (KB - 1);
      ra[i] = ((m0 + m) < M) ? A[(long long)m * lda + k] : 0.f;
    }
    int r = e >> 7, c = e & (BN - 1);
    rb[i] = ((n0 + c) < Nc) ? B[(long long)r * ldb + (n0 + c)] : 0.f;
  }
#pragma unroll
  for (int i = 0; i < 16; i++) {
    int e = tid + i * 256;
    if (TA) { As[0][e & (BM - 1)][e >> 7] = ra[i]; }
    else    { As[0][e >> 5][e & (KB - 1)] = ra[i]; }
    Bs[0][e & (BN - 1)][e >> 7] = rb[i];
  }
  __syncthreads();

  for (int ch = 0; ch < nch; ch++) {
    const int cur = ch & 1;
    const bool more = (ch + 1) < nch;

    // ---- prefetch next chunk into registers (global, overlaps WMMA below) ----
    if (more) {
      const int k0 = (ch + 1) * KB;
#pragma unroll
      for (int i = 0; i < 16; i++) {
        int e = tid + i * 256;
        if (TA) {
          int m = e & (BM - 1), k = e >> 7;
          ra[i] = ((m0 + m) < M) ? A[(long long)(k0 + k) * lda + (m0 + m)] : 0.f;
        } else {
          int m = e >> 5, k = e & (KB - 1);
          ra[i] = ((m0 + m) < M) ? A[(long long)m * lda + (k0 + k)] : 0.f;
        }
        int r = e >> 7, c = e & (BN - 1);
        rb[i] = ((n0 + c) < Nc) ? B[(long long)(k0 + r) * ldb + (n0 + c)] : 0.f;
      }
    }

    // ---- compute from LDS buf[cur] ----
    if (active) {
      const int am = wmL + (lane & 15);
      const int bn = wnL + (lane & 15);
      const int kb = (lane >> 4) << 1;  // 0 or 2
#pragma unroll
      for (int kk = 0; kk < KB; kk += 4) {
        const int ak = kk + kb;
        v2f a[4], b[2];
#pragma unroll
        for (int i = 0; i < 4; i++)
          a[i] = *(const v2f*)&As[cur][am + 16 * i][ak];
#pragma unroll
        for (int j = 0; j < 2; j++)
          b[j] = *(const v2f*)&Bs[cur][bn + 16 * j][ak];
#pragma unroll
        for (int i = 0; i < 4; i++)
#pragma unroll
          for (int j = 0; j < 2; j++) acc[i][j] = wmma4(a[i], b[j], acc[i][j]);
      }
    }

    // ---- store prefetched regs into the other buffer ----
    if (more) {
      const int nxt = cur ^ 1;
#pragma unroll
      for (int i = 0; i < 16; i++) {
        int e = tid + i * 256;
        if (TA) { As[nxt][e & (BM - 1)][e >> 7] = ra[i]; }
        else    { As[nxt][e >> 5][e & (KB - 1)] = ra[i]; }
        Bs[nxt][e & (BN - 1)][e >> 7] = rb[i];
      }
    }
    __syncthreads();
  }

  // ---- epilogue ----
  float lsum = 0.f;
  if (active) {
    const int lhi = (lane >> 4) * 8;   // C/D: vgpr r -> M=r (lanes 0-15), M=r+8 (16-31)
    const int ln  = lane & 15;
#pragma unroll
    for (int i = 0; i < 4; i++)
#pragma unroll
      for (int j = 0; j < 2; j++)
#pragma unroll
        for (int r = 0; r < 8; r++) {
          int gm = m0 + wmL + 16 * i + r + lhi;
          int gn = n0 + wnL + 16 * j + ln;
          if (gm < M && gn < Nc) {
            float v = s * acc[i][j][r];
            if (LOSS) {
              float rr = v - Adj[(long long)gm * ldadj + gn];
              float e  = expf(-rr * rr) * rr;
              lsum += e * e;
            } else {
              if (diag_ptr && gm == gn) v += dv;
              D[(long long)gm * ldd + gn] = v;
            }
          }
        }
  }
  if (LOSS) {
    red[tid] = lsum;
    __syncthreads();
    for (int st = 128; st > 0; st >>= 1) {
      if (tid < st) red[tid] += red[tid + st];
      __syncthreads();
    }
    if (tid == 0) loss_part[blockIdx.y * gridDim.x + blockIdx.x] = red[0];
  }
}

// ================= small VALU kernels =================

// query/key projection: q[l,j] = preds[l,:]·Wq[j,:] + bq[j]   ([N,64]@[64,256])
__global__ void proj_qk(const float* __restrict__ preds,
                        const float* __restrict__ Wq, const float* __restrict__ bq,
                        const float* __restrict__ Wk, const float* __restrict__ bk,
                        float* __restrict__ q, float* __restrict__ k) {
  int t = blockIdx.x * blockDim.x + threadIdx.x;
  if (t >= NNODES * NHC) return;
  int row = t >> 8, col = t & 255;
  const float* p  = preds + (long long)row * NC;
  const float* wq = Wq + (long long)col * NC;
  const float* wk = Wk + (long long)col * NC;
  float aq = bq[col], ak = bk[col];
#pragma unroll
  for (int c = 0; c < NC; c++) {
    float pv = p[c];
    aq = fmaf(pv, wq[c], aq);
    ak = fmaf(pv, wk[c], ak);
  }
  q[t] = aq;
  k[t] = ak;
}

// per-column sums + global sum-of-squares partials (grid 96, block 256, 32 rows/block)
__global__ void colreduce(const float* __restrict__ q, const float* __restrict__ k,
                          float* __restrict__ part_qsum, float* __restrict__ part_ksum,
                          float* __restrict__ part_sq) {
  int b = blockIdx.x, t = threadIdx.x;
  long long base = (long long)b * 32 * NHC;
  float qs = 0, ks = 0, q2 = 0, k2 = 0;
  for (int r = 0; r < 32; r++) {
    float qv = q[base + (long long)r * NHC + t];
    float kv = k[base + (long long)r * NHC + t];
    qs += qv; ks += kv; q2 += qv * qv; k2 += kv * kv;
  }
  part_qsum[b * NHC + t] = qs;
  part_ksum[b * NHC + t] = ks;
  __shared__ float red[256];
  red[t] = q2; __syncthreads();
  for (int st = 128; st > 0; st >>= 1) { if (t < st) red[t] += red[t + st]; __syncthreads(); }
  if (t == 0) part_sq[b * 2] = red[0];
  __syncthreads();
  red[t] = k2; __syncthreads();
  for (int st = 128; st > 0; st >>= 1) { if (t < st) red[t] += red[t + st]; __syncthreads(); }
  if (t == 0) part_sq[b * 2 + 1] = red[0];
}

// deterministic final reduce: qsum/ksum per column + 1/||q||, 1/||k||
__global__ void finalize_reduce(const float* __restrict__ part_qsum,
                                const float* __restrict__ part_ksum,
                                const float* __restrict__ part_sq, int nb,
                                float* __restrict__ qsum_raw, float* __restrict__ ksum_raw,
                                float* __restrict__ scal) {
  int t = threadIdx.x;
  float qs = 0, ks = 0;
  for (int b = 0; b < nb; b++) { qs += part_qsum[b * NHC + t]; ks += part_ksum[b * NHC + t]; }
  qsum_raw[t] = qs;
  ksum_raw[t] = ks;
  if (t == 0) {
    float sq = 0, sk = 0;
    for (int b = 0; b < nb; b++) { sq += part_sq[2 * b]; sk += part_sq[2 * b + 1]; }
    scal[0] = 1.0f / sqrtf(sq);   // qinv
    scal[1] = 1.0f / sqrtf(sk);   // kinv
  }
}

// column sums of a [N,64] matrix -> 24 partial blocks (grid 24, block 256)
__global__ void colsum_part(const float* __restrict__ src, float* __restrict__ part) {
  int b = blockIdx.x, t = threadIdx.x;
  int d = t & 63, s = t >> 6;
  long long base = (long long)b * 128 * NC;
  float a = 0;
  for (int r = 0; r < 32; r++) a += src[base + (long long)(s * 32 + r) * NC + d];
  __shared__ float red[256];
  red[t] = a; __syncthreads();
  if (s == 0) part[b * NC + d] = red[d] + red[64 + d] + red[128 + d] + red[192 + d];
}

// attention-diffusion combine: tmp_next = mean_h((qinv*q@hmd2 + colsum)/normzr);
// preds_attn += beta[bidx]*tmp_next.   grid N, block 64.
__global__ void attn_combine(const float* __restrict__ query, const float* __restrict__ hmd2,
                             const float* __restrict__ part_colsum, int ncs,
                             const float* __restrict__ ksum_raw, const float* __restrict__ scal,
                             const float* __restrict__ beta, int bidx,
                             float* __restrict__ tmp_next, float* __restrict__ preds_attn) {
  __shared__ float qrow[NHC];
  int n = blockIdx.x, d = threadIdx.x;
  for (int i = d; i < NHC; i += 64) qrow[i] = query[(long long)n * NHC + i];
  __syncthreads();
  float qinv = scal[0], kinv = scal[1], qk = qinv * kinv;
  float cs = 0;
  for (int b = 0; b < ncs; b++) cs += part_colsum[b * NC + d];
  float acc = 0;
  for (int h = 0; h < NH; h++) {
    float nz = 0;
#pragma unroll
    for (int m = 0; m < NC; m++) nz = fmaf(qrow[h * NC + m], ksum_raw[h * NC + m], nz);
    nz = qk * nz + (float)NNODES;
    float v = 0;
    const float* hp = hmd2 + h * NC * NC + d;
#pragma unroll
    for (int m = 0; m < NC; m++) v = fmaf(qrow[h * NC + m], hp[m * NC], v);
    acc += (qinv * v + cs) / nz;
  }
  acc *= 1.0f / NH;
  tmp_next[(long long)n * NC + d] = acc;
  preds_attn[(long long)n * NC + d] += beta[bidx] * acc;
}

// attn_qkq = (qinv*q@hmdq + qsum_raw)/normzr   grid N, block 256.
__global__ void qkq_finalize(const float* __restrict__ query, const float* __restrict__ hmdq,
                             const float* __restrict__ qsum_raw, const float* __restrict__ ksum_raw,
                             const float* __restrict__ scal, float* __restrict__ qkq) {
  __shared__ float qrow[NHC];
  int n = blockIdx.x, t = threadIdx.x;
  qrow[t] = query[(long long)n * NHC + t];
  __syncthreads();
  int h = t >> 6, d = t & 63;
  float qinv = scal[0], kinv = scal[1];
  float nz = 0;
#pragma unroll
  for (int m = 0; m < NC; m++) nz = fmaf(qrow[h * NC + m], ksum_raw[h * NC + m], nz);
  nz = qinv * kinv * nz + (float)NNODES;
  float v = 0;
  const float* hp = hmdq + h * NC * NC + d;
#pragma unroll
  for (int m = 0; m < NC; m++) v = fmaf(qrow[h * NC + m], hp[m * NC], v);
  qkq[(long long)n * NHC + t] = (qinv * v + qsum_raw[t]) / nz;
}

__global__ void init_diag(float* __restrict__ G, const float* __restrict__ vptr) {
  long long t = (long long)blockIdx.x * blockDim.x + threadIdx.x;
  if (t >= (long long)NNODES * NNODES) return;
  long long row = t / NNODES, col = t - row * NNODES;
  G[t] = (row == col) ? vptr[0] : 0.0f;
}

__global__ void scale_copy(float* __restrict__ dst, const float* __restrict__ src,
                           const float* __restrict__ c, int n) {
  int t = blockIdx.x * blockDim.x + threadIdx.x;
  if (t < n) dst[t] = c[0] * src[t];
}

__global__ void axpy(float* __restrict__ dst, const float* __restrict__ src,
                     const float* __restrict__ c, int n) {
  int t = blockIdx.x * blockDim.x + threadIdx.x;
  if (t < n) dst[t] += c[0] * src[t];
}

__global__ void gather_idx(const float* __restrict__ preds, const int* __restrict__ idx,
                           float* __restrict__ out) {
  int t = blockIdx.x * blockDim.x + threadIdx.x;
  if (t >= NIDX * NC) return;
  out[t] = preds[(long long)idx[t >> 6] * NC + (t & 63)];
}

__global__ void finalize_loss(const float* __restrict__ part, int n, float* __restrict__ out) {
  float a = 0;
  for (int i = 0; i < n; i++) a += part[i];
  out[0] = sqrtf(a);
}

// ================= host orchestration =================
static void g_nn(hipStream_t s, float* D, const float* A, const float* B, int M, int Nc, int K,
                 int lda, int ldb, int ldd, const float* sp, float sc, const float* dp,
                 long long ab = 0, long long bb = 0, long long db = 0, int batch = 1) {
  dim3 g((Nc + BN - 1) / BN, (M + BM - 1) / BM, batch);
  gemm_wmma<false, false><<<g, 256, 0, s>>>(D, A, B, M, Nc, K, lda, ldb, ldd, sp, sc, dp,
                                            ab, bb, db, nullptr, 0, nullptr);
}
static void g_at(hipStream_t s, float* D, const float* A, const float* B, int M, int Nc, int K,
                 int lda, int ldb, int ldd, const float* sp, float sc,
                 long long ab = 0, long long bb = 0, long long db = 0, int batch = 1) {
  dim3 g((Nc + BN - 1) / BN, (M + BM - 1) / BM, batch);
  gemm_wmma<true, false><<<g, 256, 0, s>>>(D, A, B, M, Nc, K, lda, ldb, ldd, sp, sc, nullptr,
                                           ab, bb, db, nullptr, 0, nullptr);
}

extern "C" void kernel_launch(void* const* d_in, const int* in_sizes, int n_in,
                              void* d_out, int out_size, void* d_ws, size_t ws_size,
                              hipStream_t stream) {
  (void)in_sizes; (void)n_in; (void)out_size; (void)ws_size;
  const float* preds = (const float*)d_in[0];
  const int*   idx   = (const int*)d_in[1];
  const float* Wq_w  = (const float*)d_in[2];
  const float* Wq_b  = (const float*)d_in[3];
  const float* Wk_w  = (const float*)d_in[4];
  const float* Wk_b  = (const float*)d_in[5];
  const float* alph  = (const float*)d_in[6];
  const float* beta  = (const float*)d_in[7];
  const float* Apow  = (const float*)d_in[8];
  const float* adj   = (const float*)d_in[9];
  float* out = (float*)d_out;

  const long long NN = (long long)NNODES * NNODES;
  float* p = (float*)d_ws;
  auto alloc = [&](long long n) { float* r = p; p += n; return r; };
  float* G1       = alloc(NN);                 // graph_a ping / graph_b
  float* G2       = alloc(NN);                 // graph_a pong (final graph_a)
  float* query    = alloc((long long)NNODES * NHC);
  float* key      = alloc((long long)NNODES * NHC);
  float* qkq      = alloc((long long)NNODES * NHC);
  float* hmd_big  = alloc((long long)NHC * NNODES);
  float* newvs1   = alloc((long long)NNODES * NHC);
  float* tmpA     = alloc((long long)NNODES * NC);
  float* tmpB     = alloc((long long)NNODES * NC);
  float* pattn    = alloc((long long)NNODES * NC);   // beta-weighted attention output
  float* pfinal   = alloc((long long)NNODES * NC);   // alph-weighted PPR output
  float* hmd1     = alloc(NH * NC * NC);             // also reused for hmdq
  float* hmd2     = alloc(NH * NC * NC);
  float* part_cs  = alloc(24 * NC);
  float* part_qs  = alloc(96 * NHC);
  float* part_ks  = alloc(96 * NHC);
  float* part_sq  = alloc(96 * 2);
  float* loss_pt  = alloc(24 * 24);
  float* scal     = alloc(16);
  float* qsum_raw = alloc(NHC);
  float* ksum_raw = alloc(NHC);

  // --- projections + norms/column sums ---
  proj_qk<<<(NNODES * NHC + 255) / 256, 256, 0, stream>>>(preds, Wq_w, Wq_b, Wk_w, Wk_b, query, key);
  colreduce<<<96, 256, 0, stream>>>(query, key, part_qs, part_ks, part_sq);
  finalize_reduce<<<1, 256, 0, stream>>>(part_qs, part_ks, part_sq, 96, qsum_raw, ksum_raw, scal);

  // --- attention diffusion (3 iterations), pattn = sum_t beta[t]*Ms[t] ---
  scale_copy<<<(NNODES * NC + 255) / 256, 256, 0, stream>>>(pattn, preds, beta + 0, NNODES * NC);
  const float* src = preds;
  for (int t = 1; t < NITERA; t++) {
    float* dst = (t & 1) ? tmpA : tmpB;
    colsum_part<<<24, 256, 0, stream>>>(src, part_cs);
    // hmd1[hm,d] = kinv * key^T @ src        [256 x 3072] @ [3072 x 64]
    g_at(stream, hmd1, key, src, NHC, NC, NNODES, NHC, NC, NC, scal + 1, 1.0f);
    // newvs1[:,h,:] = qinv * query_h @ hmd1_h   (batched over heads)
    g_nn(stream, newvs1, query, hmd1, NNODES, NC, NC, NHC, NC, NHC, scal + 0, 1.0f, nullptr,
         NC, NC * NC, NC, NH);
    // hmd2_h = kinv * key_h^T @ newvs1_h     (batched over heads)
    g_at(stream, hmd2, key, newvs1, NC, NC, NNODES, NHC, NHC, NC, scal + 1, 1.0f,
         NC, NC, NC * NC, NH);
    attn_combine<<<NNODES, 64, 0, stream>>>(query, hmd2, part_cs, 24, ksum_raw, scal,
                                            beta, t, dst, pattn);
    src = dst;
  }

  // --- attn_qkq = on_attention_conv(query,key,query,1) ---
  g_at(stream, hmd1, key, query, NC, NC, NNODES, NHC, NHC, NC, scal + 1, 1.0f,
       NC, NC, NC * NC, NH);
  qkq_finalize<<<NNODES, 256, 0, stream>>>(query, hmd1, qsum_raw, ksum_raw, scal, qkq);

  // --- PPR power iteration: pfinal = sum_i alph[i] * A_pow^i @ pattn ---
  scale_copy<<<(NNODES * NC + 255) / 256, 256, 0, stream>>>(pfinal, pattn, alph + 0, NNODES * NC);
  src = pattn;
  for (int i = 1; i < NITER; i++) {
    float* dst = (i & 1) ? tmpA : tmpB;
    g_nn(stream, dst, Apow, src, NNODES, NC, NNODES, NNODES, NC, NC, nullptr, 1.0f, nullptr);
    axpy<<<(NNODES * NC + 255) / 256, 256, 0, stream>>>(pfinal, dst, alph + i, NNODES * NC);
    src = dst;
  }

  // --- graph_a polynomial (final in G2) ---
  int gdiag = (int)((NN + 255) / 256);
  init_diag<<<gdiag, 256, 0, stream>>>(G1, alph + 3);
  g_nn(stream, G2, Apow, G1, NNODES, NNODES, NNODES, NNODES, NNODES, NNODES, nullptr, 1.0f, alph + 2);
  g_nn(stream, G1, Apow, G2, NNODES, NNODES, NNODES, NNODES, NNODES, NNODES, nullptr, 1.0f, alph + 1);
  g_nn(stream, G2, Apow, G1, NNODES, NNODES, NNODES, NNODES, NNODES, NNODES, nullptr, 1.0f, alph + 0);

  // --- graph_b recurrence (in G1, in-place via hmd_big) ---
  init_diag<<<gdiag, 256, 0, stream>>>(G1, beta + 3);
  for (int i = NITERA - 1; i >= 1; i--) {
    // hmd_big[hm,j] = key^T @ G1    [256 x 3072] @ [3072 x 3072]
    g_at(stream, hmd_big, key, G1, NHC, NNODES, NNODES, NHC, NNODES, NNODES, nullptr, 1.0f);
    // G1 = beta[i-1]*I + 0.25 * qkq @ hmd_big   (reads only qkq/hmd_big -> in-place OK)
    g_nn(stream, G1, qkq, hmd_big, NNODES, NNODES, NHC, NHC, NNODES, NNODES, nullptr,
         1.0f / NH, beta + (i - 1));
  }

  // --- fused loss: ||exp(-(G2@G1 - adj)^2) * (G2@G1 - adj)||_F ---
  {
    dim3 g(NNODES / BN, NNODES / BM, 1);
    gemm_wmma<false, true><<<g, 256, 0, stream>>>(nullptr, G2, G1, NNODES, NNODES, NNODES,
                                                  NNODES, NNODES, 0, nullptr, 1.0f, nullptr,
                                                  0, 0, 0, adj, NNODES, loss_pt);
  }
  finalize_loss<<<1, 1, 0, stream>>>(loss_pt, 24 * 24, out + NIDX * NC);

  // --- gather preds[idx] ---
  gather_idx<<<(NIDX * NC + 255) / 256, 256, 0, stream>>>(pfinal, idx, out);
}